// rope_78408922956139
// MI455X (gfx1250) — compile-verified
//
#include <hip/hip_runtime.h>
#include <math.h>
#include <stdint.h>

// RoPE: x (B,H,S,D=128) fp32, token_positions (S,) int32 -> out (B,H,S,128) fp32.
//
// MI455X reasoning: 268 MB in + 268 MB out @ 23.3 TB/s ~= 23 us -> pure HBM
// stream (0.25 flop/byte). No matrix contraction, so WMMA does not apply.
// Optimization = saturate HBM with 512B wave32-coalesced b128 transactions and
// amortize all transcendentals (one 64-entry cos/sin table per position s,
// shared by the B*H=128 rows of that s).
//
// CDNA5 data path: each wave double-buffers its row stream through LDS using
// GLOBAL_LOAD_ASYNC_TO_LDS_B128 (ASYNCcnt), overlapping the next row's fetch
// with the current row's compute + non-temporal store. Same HBM traffic as the
// direct path; LDS bandwidth is ample.

#define D_HEAD 128
#define D_HALF 64
#define BLOCK_THREADS 256
#define NWAVES (BLOCK_THREADS / 32)

typedef float v4f __attribute__((ext_vector_type(4)));

// Issue: copy 16 bytes per lane from global (x + byteOff) to the LDS address in ldsAddr.
// s_wait_dscnt 0x0 first: retire prior ds_loads of the slot being overwritten
// (wave ds reads and async LDS writes are NOT mutually ordered).
__device__ __forceinline__ void async_row_fetch(unsigned ldsAddr, unsigned byteOff,
                                                const float* __restrict__ base) {
  asm volatile(
      "s_wait_dscnt 0x0\n\t"
      "global_load_async_to_lds_b128 %0, %1, %2\n\t"
      :
      : "v"(ldsAddr), "v"(byteOff), "s"(base)
      : "memory");
}

__global__ __launch_bounds__(BLOCK_THREADS) void rope_gfx1250_kernel(
    const float* __restrict__ x,
    const int* __restrict__ token_positions,
    float* __restrict__ out,
    int S, int BH) {
  __shared__ float lds_cos[D_HALF];
  __shared__ float lds_sin[D_HALF];
  __shared__ v4f   stage[2][BLOCK_THREADS];  // 2 x 512B per wave, 8KB per block

  const int s   = blockIdx.x;
  const int tid = threadIdx.x;

  // --- Per-position cos/sin table (64 sincos per block, shared by 128 rows) ---
  if (tid < D_HALF) {
    const float p = (float)token_positions[s];
    // inv_theta = 10000^(-2k/128) = exp2(k * (-2/128) * log2(10000))
    const float kc = (-2.0f / 128.0f) * 13.287712379549449f;  // log2(10000)
    const float inv_theta = exp2f((float)tid * kc);
    float sv, cv;
    sincosf(p * inv_theta, &sv, &cv);  // precise range reduction (up to ~4095 rad)
    lds_cos[tid] = cv;
    lds_sin[tid] = sv;
  }
  __syncthreads();

  const int lane = tid & 31;
  const int wave = tid >> 5;

  // Lane l owns elements [4l, 4l+4) of a row: rotation pairs k=2l, 2l+1.
  const float c0 = lds_cos[2 * lane + 0];
  const float s0 = lds_sin[2 * lane + 0];
  const float c1 = lds_cos[2 * lane + 1];
  const float s1 = lds_sin[2 * lane + 1];

  const long long rowStride = (long long)S * D_HEAD;            // elems between (b,h) planes
  const long long sOff      = (long long)s * D_HEAD + lane * 4; // lane slot within row s
  const long long iterStep  = (long long)NWAVES * rowStride;

  // 32-bit LDS byte addresses of this thread's two staging slots
  // (AS3 offset == low 32 bits of the generic pointer on AMDGPU).
  const unsigned ldsSlot[2] = {
      (unsigned)(uintptr_t)&stage[0][tid],
      (unsigned)(uintptr_t)&stage[1][tid]};

  // Prologue: start async fetch of this wave's first row into slot 0.
  {
    const long long off0 = (long long)wave * rowStride + sOff;
    async_row_fetch(ldsSlot[0], (unsigned)(off0 * sizeof(float)), x);
  }

  int buf = 0;
  for (int bh = wave; bh < BH; bh += NWAVES) {
    const long long off = (long long)bh * rowStride + sOff;

    // Issue next row into the other slot (clamped on the tail: re-fetching the
    // current row keeps the async stream uniform and avoids OOB global reads).
    {
      int nbh = bh + NWAVES;
      if (nbh >= BH) nbh = bh;
      const long long noff = (long long)nbh * rowStride + sOff;
      async_row_fetch(ldsSlot[buf ^ 1], (unsigned)(noff * sizeof(float)), x);
    }

    // Warm L2 two rows ahead (gfx1250 global_prefetch_b8; speculative, OOB-safe).
    __builtin_prefetch(x + off + 2 * iterStep, 0, 0);

    // Oldest async transfer (current slot) complete when ASYNCcnt <= 1.
    asm volatile("s_wait_asynccnt 0x1" ::: "memory");

    const v4f v = stage[buf][tid];  // ds_load_b128

    v4f r;
    r.x = c0 * v.x - s0 * v.y;  // even: cos*xe - sin*xo
    r.y = s0 * v.x + c0 * v.y;  // odd:  sin*xe + cos*xo
    r.z = c1 * v.z - s1 * v.w;
    r.w = s1 * v.z + c1 * v.w;

    // Output is never re-read: non-temporal 128-bit store straight from regs.
    __builtin_nontemporal_store(r, reinterpret_cast<v4f*>(out + off));

    buf ^= 1;
  }
}

extern "C" void kernel_launch(void* const* d_in, const int* in_sizes, int n_in,
                              void* d_out, int out_size, void* d_ws, size_t ws_size,
                              hipStream_t stream) {
  const float* x   = (const float*)d_in[0];
  const int*   pos = (const int*)d_in[1];
  float*       out = (float*)d_out;

  const int S  = in_sizes[1];                 // 4096
  const int BH = in_sizes[0] / (S * D_HEAD);  // 4*32 = 128

  rope_gfx1250_kernel<<<S, BLOCK_THREADS, 0, stream>>>(x, pos, out, S, BH);
}